// RocketNet_66022237274261
// MI455X (gfx1250) — compile-verified
//
#include <hip/hip_runtime.h>

#define NKER   512
#define SLEN   2048
#define NSAMP  64
#define MAXKS  11
#define MAX_TILES 128
#define MAX_KEYS  64

// ======================================================================
// Compile-time replication of numpy RandomState(42) -> CONFIGS
// (MT19937 + legacy masked-rejection randint on 64-bit draws + rk_double)
// ======================================================================
struct MT19937 {
  unsigned int mt[624];
  int mti;
  constexpr MT19937(unsigned int seed) : mt{}, mti(624) {
    mt[0] = seed;
    for (int i = 1; i < 624; ++i)
      mt[i] = 1812433253u * (mt[i - 1] ^ (mt[i - 1] >> 30)) + (unsigned int)i;
  }
  constexpr unsigned int next32() {
    if (mti >= 624) {
      for (int i = 0; i < 624; ++i) {
        unsigned int y = (mt[i] & 0x80000000u) | (mt[(i + 1) % 624] & 0x7fffffffu);
        unsigned int v = mt[(i + 397) % 624] ^ (y >> 1);
        if (y & 1u) v ^= 0x9908b0dfu;
        mt[i] = v;
      }
      mti = 0;
    }
    unsigned int y = mt[mti++];
    y ^= (y >> 11);
    y ^= (y << 7)  & 0x9d2c5680u;
    y ^= (y << 15) & 0xefc60000u;
    y ^= (y >> 18);
    return y;
  }
  constexpr unsigned long long next64() {
    unsigned long long hi = next32();
    unsigned long long lo = next32();
    return (hi << 32) | lo;
  }
  // numpy legacy randint(0, rng+1): masked rejection on 64-bit draws
  constexpr unsigned long long bounded(unsigned long long rng) {
    unsigned long long mask = rng;
    mask |= mask >> 1;  mask |= mask >> 2;  mask |= mask >> 4;
    mask |= mask >> 8;  mask |= mask >> 16; mask |= mask >> 32;
    unsigned long long v = next64() & mask;
    while (v > rng) v = next64() & mask;
    return v;
  }
  // numpy legacy random_sample (rk_double)
  constexpr double rand01() {
    unsigned long long a = next32() >> 5, b = next32() >> 6;
    return ((double)a * 67108864.0 + (double)b) / 9007199254740992.0;
  }
};

struct CfgArr {
  int ks[NKER]; int dil[NKER]; int pad[NKER]; int off[NKER]; int olen[NKER];
  int total;
};

constexpr CfgArr make_cfgs() {
  CfgArr c{};
  MT19937 g(42u);
  const int kss[3]    = {7, 9, 11};
  const int maxdil[3] = {8, 7, 7};  // int(log2(2047/(ks-1))) for ks=7,9,11
  for (int i = 0; i < NKER; ++i) {
    int ci  = (int)g.bounded(2ull);                                  // choice of 3
    int ks  = kss[ci];
    int de  = (int)g.bounded((unsigned long long)(maxdil[ci] - 1));  // randint(0,maxdil)
    int dil = 1 << de;
    double r = g.rand01();
    int pad = (r <= 0.5) ? 0 : ((ks - 1) * dil / 2);
    c.ks[i] = ks; c.dil[i] = dil; c.pad[i] = pad;
    c.olen[i] = SLEN + 2 * pad - dil * (ks - 1);
  }
  int acc = 0;
  for (int i = 0; i < NKER; ++i) { c.off[i] = acc; acc += c.olen[i]; }
  c.total = acc;
  return c;
}
constexpr CfgArr CFG = make_cfgs();
static_assert(CFG.total > 0, "bad config totals");

// ---- Config tiles: up to 16 kernels sharing (ks, dil, pad) per WMMA tile ----
struct Tiles {
  int n;
  int ks[MAX_TILES]; int dil[MAX_TILES]; int pad[MAX_TILES]; int olen[MAX_TILES];
  int idx[MAX_TILES][16];   // global kernel index, -1 = empty row
  int off[MAX_TILES][16];   // output column offset per row
};

constexpr Tiles make_tiles() {
  Tiles T{};
  long long keyv[NKER] = {};
  for (int i = 0; i < NKER; ++i)
    keyv[i] = ((long long)CFG.ks[i] << 40) | ((long long)CFG.dil[i] << 20) |
              (long long)CFG.pad[i];
  long long keys[MAX_KEYS] = {};
  int nk = 0;
  for (int i = 0; i < NKER; ++i) {
    bool found = false;
    for (int k = 0; k < nk; ++k)
      if (keys[k] == keyv[i]) { found = true; break; }
    if (!found) keys[nk++] = keyv[i];   // OOB write -> constexpr error if too many
  }
  int nt = 0;
  for (int k = 0; k < nk; ++k) {
    int slot = 16;
    for (int i = 0; i < NKER; ++i) {
      if (keyv[i] != keys[k]) continue;
      if (slot == 16) {                 // open new tile (OOB -> constexpr error)
        T.ks[nt] = CFG.ks[i]; T.dil[nt] = CFG.dil[i];
        T.pad[nt] = CFG.pad[i]; T.olen[nt] = CFG.olen[i];
        for (int m = 0; m < 16; ++m) { T.idx[nt][m] = -1; T.off[nt][m] = 0; }
        ++nt; slot = 0;
      }
      T.idx[nt - 1][slot] = i;
      T.off[nt - 1][slot] = CFG.off[i];
      ++slot;
    }
  }
  T.n = nt;
  return T;
}
constexpr Tiles TILES = make_tiles();
static_assert(TILES.n >= 1 && TILES.n <= MAX_TILES, "tile capacity");

__constant__ Tiles c_tiles = TILES;

// ======================================================================
// Kernel: one block = (config-tile, sample). Stage the 8 KB signal row
// into LDS once (b128 loads + ds_store_b128), then 8 waves stride over
// 16-wide t-tiles computing
//   D[m][t] = sum_k W[idx_m][k] * sig[n, t0+t + k*dil - pad]  (+ bias)
// via V_WMMA_F32_16X16X4_F32, K padded to 12 (3 K-steps of 4),
// B operands fed from LDS (lanes 0-15 hit consecutive banks).
// ======================================================================
typedef float v2f __attribute__((ext_vector_type(2)));
typedef float v4f __attribute__((ext_vector_type(4)));
typedef float v8f __attribute__((ext_vector_type(8)));

__global__ __launch_bounds__(256) void rocket_conv_wmma(
    const float* __restrict__ sig, const float* __restrict__ W,
    const float* __restrict__ b, float* __restrict__ out, int total_len) {
  __shared__ float s_sig[SLEN];   // 8 KB: this sample's signal row

  const int tile = blockIdx.x;
  const int n    = blockIdx.y;
  const int lane = threadIdx.x & 31;
  const int wave = threadIdx.x >> 5;
  const int row  = lane & 15;   // N column (B/D) and M row (A)
  const int hi   = lane >> 4;   // upper half-wave flag

  const int ks   = c_tiles.ks[tile];
  const int dil  = c_tiles.dil[tile];
  const int pad  = c_tiles.pad[tile];
  const int olen = c_tiles.olen[tile];
  const int* tidx = c_tiles.idx[tile];
  const int* toff = c_tiles.off[tile];

  const float* __restrict__ srow = sig + n * SLEN;
  float* __restrict__ orow = out + (size_t)n * (size_t)total_len;

  // ---- stage signal row to LDS: 512 x b128, 2 per thread ----
  {
    const v4f* __restrict__ src = (const v4f*)srow;
    v4f* dst = (v4f*)s_sig;
#pragma unroll
    for (int i = threadIdx.x; i < SLEN / 4; i += 256) dst[i] = src[i];
  }

  // ---- A operand (loop-invariant): 16 kernel rows x 12 taps of f32 ----
  // A 16x4 layout: VGPR0 = K(2*hi), VGPR1 = K(2*hi+1), M = lane&15.
  const int kq = tidx[row];
  const float* wrow = W + (kq < 0 ? 0 : kq) * MAXKS;
  v2f A[3];
#pragma unroll
  for (int s = 0; s < 3; ++s) {
    const int k0 = 4 * s + 2 * hi;
    float ax = (kq >= 0 && k0     < ks) ? wrow[k0]     : 0.0f;
    float ay = (kq >= 0 && k0 + 1 < ks) ? wrow[k0 + 1] : 0.0f;
    A[s].x = ax; A[s].y = ay;
  }

  // ---- store metadata: D VGPR v holds row m = v + 8*hi, col = row ----
  int   st_off[8];
  float st_b[8];
  int   st_ok[8];
#pragma unroll
  for (int v = 0; v < 8; ++v) {
    const int m  = v + 8 * hi;
    const int ki = tidx[m];
    st_ok[v]  = (ki >= 0);
    st_off[v] = toff[m];
    st_b[v]   = (ki >= 0) ? b[ki] : 0.0f;
  }

  __syncthreads();   // LDS staging visible to all waves

  const int ksteps = (ks + 3) >> 2;        // 2 for ks=7, 3 for ks=9/11 (uniform)
  const int ntt    = (olen + 15) >> 4;
  for (int tt = wave; tt < ntt; tt += 8) {
    const int t0 = tt << 4;
    v8f c = {0.f, 0.f, 0.f, 0.f, 0.f, 0.f, 0.f, 0.f};
    const int pb = t0 + row - pad;
    for (int s = 0; s < ksteps; ++s) {
      // B 4x16 layout mirrors A: VGPR0 = K(2*hi), VGPR1 = K(2*hi+1), N = lane&15
      int p0 = pb + (4 * s + 2 * hi) * dil;
      int p1 = p0 + dil;
      int q0 = p0 < 0 ? 0 : (p0 > SLEN - 1 ? SLEN - 1 : p0);
      int q1 = p1 < 0 ? 0 : (p1 > SLEN - 1 ? SLEN - 1 : p1);
      float bx = s_sig[q0]; bx = (q0 == p0) ? bx : 0.0f;  // branchless edge zeroing
      float by = s_sig[q1]; by = (q1 == p1) ? by : 0.0f;
      v2f B; B.x = bx; B.y = by;
      // (neg_a, A, neg_b, B, c_mod, C, reuse_a, reuse_b)
      c = __builtin_amdgcn_wmma_f32_16x16x4_f32(false, A[s], false, B,
                                                (short)0, c, false, false);
    }
    const int tcol = t0 + row;
    if (tcol < olen) {
#pragma unroll
      for (int v = 0; v < 8; ++v) {
        if (st_ok[v]) orow[st_off[v] + tcol] = c[v] + st_b[v];
      }
    }
  }
}

extern "C" void kernel_launch(void* const* d_in, const int* in_sizes, int n_in,
                              void* d_out, int out_size, void* d_ws, size_t ws_size,
                              hipStream_t stream) {
  (void)in_sizes; (void)n_in; (void)d_ws; (void)ws_size; (void)out_size;
  const float* sig = (const float*)d_in[0];   // (64, 2048) f32
  const float* W   = (const float*)d_in[1];   // (512, 11) f32
  const float* b   = (const float*)d_in[2];   // (512,) f32
  float* out = (float*)d_out;                 // (64, 1, total_len) f32

  dim3 grid((unsigned)TILES.n, NSAMP, 1);
  rocket_conv_wmma<<<grid, 256, 0, stream>>>(sig, W, b, out, CFG.total);
}